// TCGroupPoolingR2Spatial_18614388261224
// MI455X (gfx1250) — compile-verified
//
#include <hip/hip_runtime.h>

typedef __attribute__((ext_vector_type(2))) float v2f;
typedef __attribute__((ext_vector_type(8))) float v8f;

#define D 256
#define AS_STRIDE 260            // 260 % 64 == 4 -> conflict-free K-striped A reads
#define AS_SIZE   (32 * AS_STRIDE)   // 8320 dwords, 8320 % 64 == 0
#define XD_OFF    AS_SIZE            // xd  base bank  0 (mod 64)
#define XD2_OFF   (AS_SIZE + 512 + 14) // xd2 base bank 14 (mod 64): upper-half B
                                       // window [s+16,s+31] disjoint from lower [s,s+15]
#define LDS_DW    (XD2_OFF + 512)    // 9358 dwords = 37432 bytes

// TC_m = (C * diag(x)) * C  with C[i,j] = x[(i+j)%256]
// Block = 128 threads (4 waves), owns one (m, 32-row strip) x all 256 columns.
// Each wave: 2 a-tiles x 4 c-tiles (8 accumulators) -> 1.5 LDS dwords per WMMA.
__global__ __launch_bounds__(128) void tc_circulant_wmma_kernel(
    const float* __restrict__ x, float* __restrict__ out, int n_m)
{
    __shared__ float lds[LDS_DW];
    float* As  = lds;              // As[r][i] = x[i]*x[a0+r+i], r in [0,32)
    float* xd  = lds + XD_OFF;     // x duplicated (wrap-free circulant), lanes 0-15 B
    float* xd2 = lds + XD2_OFF;    // staggered duplicate, lanes 16-31 B (bank-disjoint)

    const int tid = threadIdx.x;
    const int blk = blockIdx.x;
    const int m     = blk >> 3;    // which of the n_m rows
    const int strip = blk & 7;     // which 32-row strip of the 256x256 output
    const int a0    = strip * 32;

    const float* xm = x + (size_t)m * D;

    // Stage x into both duplicated LDS copies (idx%256 becomes a plain add, idx<512).
    for (int i = tid; i < 2 * D; i += 128) {
        const float v = xm[i & (D - 1)];
        xd[i]  = v;
        xd2[i] = v;
    }
    __syncthreads();

    // Build the shared A strip for this block's 32 output rows:
    // As[r][i] = x[i] * x[(a0 + r + i) % 256]   (diag(x) folded into A)
    {
        const int r  = tid & 31;          // row 0..31
        const int i0 = (tid >> 5) * 64;   // 4 chunks of 64 columns
        #pragma unroll
        for (int j = 0; j < 64; ++j) {
            const int i = i0 + j;
            As[r * AS_STRIDE + i] = xd[i] * xd[a0 + r + i];
        }
    }
    __syncthreads();

    const int wave  = tid >> 5;        // 0..3  -> this wave's quad of c-tiles
    const int lane  = tid & 31;
    const int lrow  = lane & 15;       // M for A-reads, N for B-reads
    const int uh    = lane >> 4;       // upper half-wave flag
    const int khalf = uh * 2;          // K sub-pair: lanes 0-15 K{0,1}, 16-31 K{2,3}

    const int c0 = wave * 64 + lrow;   // 4 c-tiles: c0, c0+16, c0+32, c0+48

    v8f acc[2][4];
    #pragma unroll
    for (int p = 0; p < 2; ++p)
        #pragma unroll
        for (int q = 0; q < 4; ++q)
            acc[p][q] = (v8f){};

    const float* A0 = &As[lrow * AS_STRIDE];          // rows a0 + 0..15
    const float* A1 = A0 + 16 * AS_STRIDE;            // rows a0 + 16..31
    const float* bp = &lds[XD_OFF + uh * (XD2_OFF - XD_OFF)]; // half-wave B copy

    // K = 256 in 64 steps of V_WMMA_F32_16X16X4_F32 (8 wmma per step)
    #pragma unroll 2
    for (int k0 = 0; k0 < D; k0 += 4) {
        const int kb = k0 + khalf;
        v2f a0v, a1v, b[4];
        // A 16x4 layout (ISA 7.12.2): lane M=lrow holds K = kb, kb+1
        a0v.x = A0[kb];
        a0v.y = A0[kb + 1];
        a1v.x = A1[kb];
        a1v.y = A1[kb + 1];
        // B 4x16 layout: lane N holds rows K = kb, kb+1; B[k][c] = x[k+c]
        #pragma unroll
        for (int q = 0; q < 4; ++q) {
            b[q].x = bp[kb + c0 + q * 16];
            b[q].y = bp[kb + 1 + c0 + q * 16];
        }
        // 8-arg form: (neg_a, A, neg_b, B, c_mod, C, reuse_a, reuse_b)
        #pragma unroll
        for (int q = 0; q < 4; ++q) {
            acc[0][q] = __builtin_amdgcn_wmma_f32_16x16x4_f32(
                false, a0v, false, b[q], (short)0, acc[0][q], false, false);
            acc[1][q] = __builtin_amdgcn_wmma_f32_16x16x4_f32(
                false, a1v, false, b[q], (short)0, acc[1][q], false, false);
        }
    }

    // D 16x16 layout: VGPR r -> output row r (lanes 0-15) / r+8 (lanes 16-31)
    const int half8 = uh * 8;
    float* op = out + (size_t)m * (D * D) + (size_t)a0 * D;
    #pragma unroll
    for (int r = 0; r < 8; ++r) {
        const int ar = r + half8;
        #pragma unroll
        for (int p = 0; p < 2; ++p) {
            #pragma unroll
            for (int q = 0; q < 4; ++q) {
                op[(size_t)(ar + p * 16) * D + c0 + q * 16] = acc[p][q][r];
            }
        }
    }
}

extern "C" void kernel_launch(void* const* d_in, const int* in_sizes, int n_in,
                              void* d_out, int out_size, void* d_ws, size_t ws_size,
                              hipStream_t stream) {
    (void)n_in; (void)out_size; (void)d_ws; (void)ws_size;
    const float* x = (const float*)d_in[0];
    float* out = (float*)d_out;

    const int n_m = in_sizes[0] / D;     // b*k = 1024 rows of length 256
    dim3 grid((unsigned)(n_m * 8));      // 8 strips of 32 rows per m
    dim3 block(128);                     // 4 waves; EXEC all-ones for WMMA
    tc_circulant_wmma_kernel<<<grid, block, 0, stream>>>(x, out, n_m);
}